// EdgeDotProductMPN_50414326121237
// MI455X (gfx1250) — compile-verified
//
#include <hip/hip_runtime.h>
#include <hip/hip_bf16.h>
#include <stdint.h>

typedef __attribute__((ext_vector_type(2))) float v2f;
typedef __attribute__((ext_vector_type(8))) float v8f;

#define DFEAT 128

__global__ __launch_bounds__(256) void EdgeDotWmma_kernel(
    const float* __restrict__ x,
    const int*   __restrict__ src,
    const int*   __restrict__ dst,
    float*       __restrict__ out,
    int n_edges, int n_groups)
{
    const int lane          = threadIdx.x & 31;
    const int waveInBlock   = threadIdx.x >> 5;
    const int wavesPerBlock = blockDim.x >> 5;
    const int waveId        = blockIdx.x * wavesPerBlock + waveInBlock;
    const int nWaves        = gridDim.x * wavesPerBlock;

    // A (16x4 f32) and B (4x16 f32) have the same load pattern from the
    // gathered-row point of view: lane l supplies row (l&15) of its matrix,
    // columns kbase + 2*(l>>4) .. +1 (an aligned float2).
    const int mrow    = lane & 15;        // edge slot within the 16-edge group
    const int colHalf = (lane >> 4) << 1; // 0 for lanes 0-15, 2 for lanes 16-31

    for (int g = waveId; g < n_groups; g += nWaves) {
        const int e0 = g << 4;
        int e = e0 + mrow;
        if (e >= n_edges) e = n_edges - 1;   // clamp tail for loads only
        const int rs = src[e];
        const int rd = dst[e];
        const float* ps = x + (size_t)rs * DFEAT + colHalf;
        const float* pd = x + (size_t)rd * DFEAT + colHalf;

        v8f c = {0.f, 0.f, 0.f, 0.f, 0.f, 0.f, 0.f, 0.f};
        #pragma unroll
        for (int k = 0; k < DFEAT; k += 4) {
            v2f a = *(const v2f*)(ps + k);   // global_load_b64, A 16x4 tile
            v2f b = *(const v2f*)(pd + k);   // global_load_b64, B 4x16 tile
            // D(16x16,f32) = A(16x4)*B(4x16) + C  — chain over K=128
            c = __builtin_amdgcn_wmma_f32_16x16x4_f32(
                    false, a, false, b, (short)0, c, false, false);
        }

        // Diagonal extraction: C[m][m] lives in VGPR (m&7) of
        //   lane m      for m = 0..7
        //   lane m+16   for m = 8..15
        const int srcLane = (lane < 8) ? lane : lane + 16;
        float r = 0.f;
        #pragma unroll
        for (int j = 0; j < 8; ++j) {
            float t = __shfl(c[j], srcLane, 32);
            if ((lane & 7) == j) r = t;
        }

        const int eo = e0 + lane;
        if (lane < 16 && eo < n_edges) out[eo] = r;
    }
}

extern "C" void kernel_launch(void* const* d_in, const int* in_sizes, int n_in,
                              void* d_out, int out_size, void* d_ws, size_t ws_size,
                              hipStream_t stream) {
    const float* x  = (const float*)d_in[0];
    const int*   ei = (const int*)d_in[1];   // edge_index, flat (2, E)
    const int n_edges = out_size;            // one output per edge
    const int* srcIdx = ei;
    const int* dstIdx = ei + n_edges;

    const int n_groups = (n_edges + 15) / 16;      // 16 edges per wave-group
    const int threads  = 256;                       // 8 wave32 per block
    const int wavesPerBlock = threads / 32;
    int blocks = (n_groups + wavesPerBlock - 1) / wavesPerBlock;
    if (blocks > 32768) blocks = 32768;             // grid-stride beyond this

    EdgeDotWmma_kernel<<<blocks, threads, 0, stream>>>(
        x, srcIdx, dstIdx, (float*)d_out, n_edges, n_groups);
}